// SpMultiHeadHigherOrderAttentionClassifier_19696720019806
// MI455X (gfx1250) — compile-verified
//
#include <hip/hip_runtime.h>
#include <math.h>

#define NCELL 50000
#define NEDGE 800000
#define DIN   128
#define FO    32
#define NH    8
#define NC    8
#define HFO   (NH * FO)      // 256
#define LALPHA 0.1f

typedef float v2f __attribute__((ext_vector_type(2)));
typedef float v8f __attribute__((ext_vector_type(8)));

__device__ __forceinline__ float lrelu(float x) { return x > 0.f ? x : LALPHA * x; }

__device__ __forceinline__ void atomicMaxF(float* addr, float val) {
    unsigned int* ua = (unsigned int*)addr;
    unsigned int old = __hip_atomic_load(ua, __ATOMIC_RELAXED, __HIP_MEMORY_SCOPE_AGENT);
    while (__uint_as_float(old) < val) {
        unsigned int assumed = old;
        old = atomicCAS(ua, assumed, __float_as_uint(val));
        if (old == assumed) break;
    }
}

// ---------------------------------------------------------------------------
// Generic f32 WMMA GEMM: C[M,Nn] = A[M,K] @ B[K,Nn].
// Requirements: M % 16 == 0, Nn % 16 == 0, K % 4 == 0. One wave per 16x16 tile.
// A fragment (16x4 f32, 2 VGPRs): lanes 0-15 -> K = k+0,k+1 ; lanes 16-31 -> K = k+2,k+3
// B fragment (4x16 f32, 2 VGPRs): same K split, column = lane%16
// D (16x16 f32, 8 VGPRs): row = v + 8*(lane/16), col = lane%16
// ---------------------------------------------------------------------------
__global__ void gemm_wmma_f32(const float* __restrict__ A, const float* __restrict__ B,
                              float* __restrict__ C, int M, int Nn, int K) {
    const int lane = threadIdx.x & 31;
    const int wave = (int)((blockIdx.x * blockDim.x + threadIdx.x) >> 5);
    const int nt = Nn >> 4;
    const int mt = M >> 4;
    if (wave >= mt * nt) return;               // wave-uniform exit: EXEC stays all-1s
    const int tm = wave / nt;
    const int tn = wave - tm * nt;
    const int half = lane >> 4;                // 0 or 1
    const int kh = half * 2;                   // K sub-offset for this half-wave
    const int rowA = tm * 16 + (lane & 15);
    const int colB = tn * 16 + (lane & 15);
    const float* ap = A + (size_t)rowA * K;

    v8f acc = {0.f, 0.f, 0.f, 0.f, 0.f, 0.f, 0.f, 0.f};
    for (int k = 0; k < K; k += 4) {
        v2f a = *(const v2f*)(ap + k + kh);    // contiguous pair, 8B aligned
        v2f b;
        b.x = B[(size_t)(k + kh) * Nn + colB];
        b.y = B[(size_t)(k + kh + 1) * Nn + colB];
        acc = __builtin_amdgcn_wmma_f32_16x16x4_f32(false, a, false, b,
                                                    (short)0, acc, false, false);
    }
#pragma unroll
    for (int v = 0; v < 8; ++v)
        C[(size_t)(tm * 16 + v + 8 * half) * Nn + colB] = acc[v];
}

// ---------------------------------------------------------------------------
// Small helper kernels
// ---------------------------------------------------------------------------
__global__ void fillk(float* p, int n, float v) {
    int i = blockIdx.x * blockDim.x + threadIdx.x;
    if (i < n) p[i] = v;
}

// Ws_h [H, DIN, FO] -> Bs [DIN, H*FO]   (Bs[k][h*32+f] = Ws_h[h][k][f])
__global__ void pack_b1(const float* __restrict__ W, float* __restrict__ Bs) {
    int idx = blockIdx.x * blockDim.x + threadIdx.x;
    if (idx >= DIN * HFO) return;
    int k = idx / HFO, c = idx - k * HFO;
    int h = c >> 5, f = c & 31;
    Bs[idx] = W[h * (DIN * FO) + k * FO + f];
}

// Wo [HFO, NC] -> B2 [HFO, 16] zero-padded cols 8..15
__global__ void pack_b2(const float* __restrict__ W, float* __restrict__ B2) {
    int idx = blockIdx.x * blockDim.x + threadIdx.x;
    if (idx >= HFO * 16) return;
    int k = idx >> 4, c = idx & 15;
    B2[idx] = (c < NC) ? W[k * NC + c] : 0.f;
}

// per-node attention scalars, layer 1 (per head)
__global__ void scores1(const float* __restrict__ hs, const float* __restrict__ ht,
                        const float* __restrict__ a1, const float* __restrict__ a2,
                        float* s1s, float* s1t, float* s2t, float* s2s) {
    int idx = blockIdx.x * blockDim.x + threadIdx.x;   // n*NH + h
    if (idx >= NCELL * NH) return;
    int n = idx >> 3, h = idx & 7;
    const float* hsp = hs + (size_t)n * HFO + h * FO;
    const float* htp = ht + (size_t)n * HFO + h * FO;
    const float* a1p = a1 + h * (2 * FO);
    const float* a2p = a2 + h * (2 * FO);
    float v1s = 0.f, v1t = 0.f, v2t = 0.f, v2s = 0.f;
#pragma unroll 8
    for (int f = 0; f < FO; ++f) {
        float xs = hsp[f], xt = htp[f];
        v1s += xs * a1p[f];
        v1t += xt * a1p[FO + f];
        v2t += xt * a2p[f];
        v2s += xs * a2p[FO + f];
    }
    s1s[idx] = v1s; s1t[idx] = v1t; s2t[idx] = v2t; s2s[idx] = v2s;
}

// layer-1 segment max (pass 1): one thread per (edge, head) -> consecutive lanes
// touch consecutive head slots of the same edge pair (coalesced per 4 edges).
__global__ void edge_max1(const int* __restrict__ et, const int* __restrict__ es,
                          const float* __restrict__ s1s, const float* __restrict__ s1t,
                          const float* __restrict__ s2t, const float* __restrict__ s2s,
                          float* m1, float* m2) {
    int idx = blockIdx.x * blockDim.x + threadIdx.x;   // e*NH + h
    if (idx >= NEDGE * NH) return;
    int e = idx >> 3, h = idx & 7;
    int t = et[e], s = es[e];
    float e1 = lrelu(s1s[s * NH + h] + s1t[t * NH + h]);
    float e2 = lrelu(s2t[t * NH + h] + s2s[s * NH + h]);
    atomicMaxF(&m1[t * NH + h], e1);
    atomicMaxF(&m2[s * NH + h], e2);
}

// layer-1 exp/denominator + weighted scatter (pass 2):
// one lane per (edge, head, feature); a full wave32 covers one (edge, head),
// so the 32-float feature gather and the 32-float atomic scatter are each a
// single contiguous 128B transaction (ideal for L2 atomic coalescing).
__global__ void edge_acc1(const int* __restrict__ et, const int* __restrict__ es,
                          const float* __restrict__ s1s, const float* __restrict__ s1t,
                          const float* __restrict__ s2t, const float* __restrict__ s2s,
                          const float* __restrict__ m1, const float* __restrict__ m2,
                          const float* __restrict__ hs, const float* __restrict__ ht,
                          float* d1, float* d2, float* accTs, float* accSt) {
    unsigned int idx = blockIdx.x * blockDim.x + threadIdx.x;  // ((e*NH)+h)*FO + f
    if (idx >= (unsigned int)NEDGE * NH * FO) return;
    int f = (int)(idx & 31);
    int h = (int)((idx >> 5) & 7);
    int e = (int)(idx >> 8);
    int t = et[e], s = es[e];
    float e1 = lrelu(s1s[s * NH + h] + s1t[t * NH + h]);
    float e2 = lrelu(s2t[t * NH + h] + s2s[s * NH + h]);
    float w1 = __expf(e1 - m1[t * NH + h]);
    float w2 = __expf(e2 - m2[s * NH + h]);
    if (f == 0) {                       // one lane owns the denominator update
        atomicAdd(&d1[t * NH + h], w1);
        atomicAdd(&d2[s * NH + h], w2);
    }
    atomicAdd(&accTs[(size_t)t * HFO + h * FO + f], w1 * hs[(size_t)s * HFO + h * FO + f]);
    atomicAdd(&accSt[(size_t)s * HFO + h * FO + f], w2 * ht[(size_t)t * HFO + h * FO + f]);
}

// layer-1 finalize: divide by denom + ELU, in place
__global__ void finalize1(float* accTs, float* accSt,
                          const float* __restrict__ d1, const float* __restrict__ d2) {
    int idx = blockIdx.x * blockDim.x + threadIdx.x;   // n*HFO + c
    if (idx >= NCELL * HFO) return;
    int n = idx / HFO, c = idx - n * HFO, h = c >> 5;
    float dd1 = d1[n * NH + h], dd2 = d2[n * NH + h];
    float v1 = dd1 > 0.f ? accTs[idx] / dd1 : 0.f;
    float v2 = dd2 > 0.f ? accSt[idx] / dd2 : 0.f;
    accTs[idx] = v1 > 0.f ? v1 : expm1f(v1);
    accSt[idx] = v2 > 0.f ? v2 : expm1f(v2);
}

// per-node attention scalars, layer 2
__global__ void scores2(const float* __restrict__ hs2, const float* __restrict__ ht2,
                        const float* __restrict__ a1o, const float* __restrict__ a2o,
                        float* s1s, float* s1t, float* s2t, float* s2s) {
    int n = blockIdx.x * blockDim.x + threadIdx.x;
    if (n >= NCELL) return;
    float v1s = 0.f, v1t = 0.f, v2t = 0.f, v2s = 0.f;
#pragma unroll
    for (int c = 0; c < NC; ++c) {
        float xs = hs2[n * 16 + c], xt = ht2[n * 16 + c];
        v1s += xs * a1o[c];
        v1t += xt * a1o[NC + c];
        v2t += xt * a2o[c];
        v2s += xs * a2o[NC + c];
    }
    s1s[n] = v1s; s1t[n] = v1t; s2t[n] = v2t; s2s[n] = v2s;
}

__global__ void edge_max2(const int* __restrict__ et, const int* __restrict__ es,
                          const float* __restrict__ s1s, const float* __restrict__ s1t,
                          const float* __restrict__ s2t, const float* __restrict__ s2s,
                          float* m1, float* m2) {
    int e = blockIdx.x * blockDim.x + threadIdx.x;
    if (e >= NEDGE) return;
    int t = et[e], s = es[e];
    atomicMaxF(&m1[t], lrelu(s1s[s] + s1t[t]));
    atomicMaxF(&m2[s], lrelu(s2t[t] + s2s[s]));
}

// layer-2 accumulate: one lane per (edge, class); 8 consecutive lanes cover one
// edge's class vector (contiguous 32B scatter); lane c==0 owns the denominator.
__global__ void edge_acc2(const int* __restrict__ et, const int* __restrict__ es,
                          const float* __restrict__ s1s, const float* __restrict__ s1t,
                          const float* __restrict__ s2t, const float* __restrict__ s2s,
                          const float* __restrict__ m1, const float* __restrict__ m2,
                          const float* __restrict__ hs2, const float* __restrict__ ht2,
                          float* d1, float* d2, float* accTs, float* accSt) {
    int idx = blockIdx.x * blockDim.x + threadIdx.x;   // e*NC + c
    if (idx >= NEDGE * NC) return;
    int c = idx & 7, e = idx >> 3;
    int t = et[e], s = es[e];
    float w1 = __expf(lrelu(s1s[s] + s1t[t]) - m1[t]);
    float w2 = __expf(lrelu(s2t[t] + s2s[s]) - m2[s]);
    if (c == 0) {
        atomicAdd(&d1[t], w1);
        atomicAdd(&d2[s], w2);
    }
    atomicAdd(&accTs[t * NC + c], w1 * hs2[s * 16 + c]);
    atomicAdd(&accSt[s * NC + c], w2 * ht2[t * 16 + c]);
}

// layer-2 finalize: divide + ELU + log_softmax. out[0:N*C] = o_st, out[N*C:] = o_ts
__global__ void finalize2(const float* __restrict__ accSt, const float* __restrict__ accTs,
                          const float* __restrict__ d2, const float* __restrict__ d1,
                          float* __restrict__ out) {
    int n = blockIdx.x * blockDim.x + threadIdx.x;
    if (n >= NCELL) return;
    // o_st (from acc over edge_s / d2)
    {
        float dd = d2[n], v[NC], m = -3.0e38f;
#pragma unroll
        for (int c = 0; c < NC; ++c) {
            float x = dd > 0.f ? accSt[n * NC + c] / dd : 0.f;
            x = x > 0.f ? x : expm1f(x);
            v[c] = x; m = fmaxf(m, x);
        }
        float sum = 0.f;
#pragma unroll
        for (int c = 0; c < NC; ++c) sum += __expf(v[c] - m);
        float lse = m + __logf(sum);
#pragma unroll
        for (int c = 0; c < NC; ++c) out[n * NC + c] = v[c] - lse;
    }
    // o_ts (from acc over edge_t / d1)
    {
        float dd = d1[n], v[NC], m = -3.0e38f;
#pragma unroll
        for (int c = 0; c < NC; ++c) {
            float x = dd > 0.f ? accTs[n * NC + c] / dd : 0.f;
            x = x > 0.f ? x : expm1f(x);
            v[c] = x; m = fmaxf(m, x);
        }
        float sum = 0.f;
#pragma unroll
        for (int c = 0; c < NC; ++c) sum += __expf(v[c] - m);
        float lse = m + __logf(sum);
#pragma unroll
        for (int c = 0; c < NC; ++c) out[(size_t)NCELL * NC + n * NC + c] = v[c] - lse;
    }
}

static inline int cdiv(long long a, int b) { return (int)((a + b - 1) / b); }

extern "C" void kernel_launch(void* const* d_in, const int* in_sizes, int n_in,
                              void* d_out, int out_size, void* d_ws, size_t ws_size,
                              hipStream_t stream) {
    const float* x_src = (const float*)d_in[0];
    const float* x_tgt = (const float*)d_in[1];
    const int*   et    = (const int*)d_in[2];
    const int*   es    = (const int*)d_in[3];
    const float* Ws_h  = (const float*)d_in[4];
    const float* Wt_h  = (const float*)d_in[5];
    const float* a1_h  = (const float*)d_in[6];
    const float* a2_h  = (const float*)d_in[7];
    const float* Ws_o  = (const float*)d_in[8];
    const float* Wt_o  = (const float*)d_in[9];
    const float* a1_o  = (const float*)d_in[10];
    const float* a2_o  = (const float*)d_in[11];
    float* out = (float*)d_out;

    // ---- workspace carve-up (all float32) ----
    float* w = (float*)d_ws;
    size_t o = 0;
    float* Bs   = w + o; o += (size_t)DIN * HFO;
    float* Bt   = w + o; o += (size_t)DIN * HFO;
    float* hsA  = w + o; o += (size_t)NCELL * HFO;
    float* htA  = w + o; o += (size_t)NCELL * HFO;
    float* s1s  = w + o; o += (size_t)NCELL * NH;
    float* s1t  = w + o; o += (size_t)NCELL * NH;
    float* s2t  = w + o; o += (size_t)NCELL * NH;
    float* s2s  = w + o; o += (size_t)NCELL * NH;
    float* m1   = w + o; o += (size_t)NCELL * NH;   // m1,m2 contiguous for fill
    float* m2   = w + o; o += (size_t)NCELL * NH;
    float* d1   = w + o; o += (size_t)NCELL * NH;   // d1,d2 contiguous
    float* d2   = w + o; o += (size_t)NCELL * NH;
    float* accTs= w + o; o += (size_t)NCELL * HFO;  // accTs,accSt contiguous
    float* accSt= w + o; o += (size_t)NCELL * HFO;
    float* B2s  = w + o; o += (size_t)HFO * 16;
    float* B2t  = w + o; o += (size_t)HFO * 16;
    float* hs2  = w + o; o += (size_t)NCELL * 16;
    float* ht2  = w + o; o += (size_t)NCELL * 16;
    float* t1s  = w + o; o += NCELL;
    float* t1t  = w + o; o += NCELL;
    float* t2t  = w + o; o += NCELL;
    float* t2s  = w + o; o += NCELL;
    float* M1   = w + o; o += NCELL;                // M1,M2 contiguous
    float* M2   = w + o; o += NCELL;
    float* D1   = w + o; o += NCELL;                // D1,D2 contiguous
    float* D2   = w + o; o += NCELL;
    float* A2ts = w + o; o += (size_t)NCELL * NC;   // A2ts,A2st contiguous
    float* A2st = w + o; o += (size_t)NCELL * NC;
    (void)ws_size; (void)in_sizes; (void)n_in; (void)out_size;

    const int BT = 256;

    // 1) pack layer-1 weight blocks
    pack_b1<<<cdiv(DIN * HFO, BT), BT, 0, stream>>>(Ws_h, Bs);
    pack_b1<<<cdiv(DIN * HFO, BT), BT, 0, stream>>>(Wt_h, Bt);

    // 2) WMMA projections: hsA = x_src @ Bs, htA = x_tgt @ Bt  [50000 x 256]
    {
        int waves = (NCELL / 16) * (HFO / 16);          // 50000 waves
        int blocks = cdiv((long long)waves * 32, BT);
        gemm_wmma_f32<<<blocks, BT, 0, stream>>>(x_src, Bs, hsA, NCELL, HFO, DIN);
        gemm_wmma_f32<<<blocks, BT, 0, stream>>>(x_tgt, Bt, htA, NCELL, HFO, DIN);
    }

    // 3) per-node attention scalars (layer 1)
    scores1<<<cdiv(NCELL * NH, BT), BT, 0, stream>>>(hsA, htA, a1_h, a2_h, s1s, s1t, s2t, s2s);

    // 4) init segment-softmax state (every call — ws is not re-zeroed by harness)
    fillk<<<cdiv(2 * NCELL * NH, BT), BT, 0, stream>>>(m1, 2 * NCELL * NH, -3.0e38f);
    fillk<<<cdiv(2 * NCELL * NH, BT), BT, 0, stream>>>(d1, 2 * NCELL * NH, 0.f);
    fillk<<<cdiv(2 * NCELL * HFO, BT), BT, 0, stream>>>(accTs, 2 * NCELL * HFO, 0.f);
    fillk<<<cdiv(2 * NCELL, BT), BT, 0, stream>>>(M1, 2 * NCELL, -3.0e38f);
    fillk<<<cdiv(2 * NCELL, BT), BT, 0, stream>>>(D1, 2 * NCELL, 0.f);
    fillk<<<cdiv(2 * NCELL * NC, BT), BT, 0, stream>>>(A2ts, 2 * NCELL * NC, 0.f);

    // 5) layer-1 segment softmax + scatter
    edge_max1<<<cdiv(NEDGE * NH, BT), BT, 0, stream>>>(et, es, s1s, s1t, s2t, s2s, m1, m2);
    {
        long long tot = (long long)NEDGE * NH * FO;     // 204.8M lanes
        edge_acc1<<<cdiv(tot, BT), BT, 0, stream>>>(et, es, s1s, s1t, s2t, s2s,
                                                    m1, m2, hsA, htA, d1, d2, accTs, accSt);
    }
    finalize1<<<cdiv(NCELL * HFO, BT), BT, 0, stream>>>(accTs, accSt, d1, d2);
    // accTs = out_ts (ELU'd), accSt = out_st

    // 6) layer-2 projections (classes padded to 16): hs2 = out_ts @ Ws_o, ht2 = out_st @ Wt_o
    pack_b2<<<cdiv(HFO * 16, BT), BT, 0, stream>>>(Ws_o, B2s);
    pack_b2<<<cdiv(HFO * 16, BT), BT, 0, stream>>>(Wt_o, B2t);
    {
        int waves = (NCELL / 16) * (16 / 16);           // 3125 waves
        int blocks = cdiv((long long)waves * 32, BT);
        gemm_wmma_f32<<<blocks, BT, 0, stream>>>(accTs, B2s, hs2, NCELL, 16, HFO);
        gemm_wmma_f32<<<blocks, BT, 0, stream>>>(accSt, B2t, ht2, NCELL, 16, HFO);
    }

    // 7) layer-2 attention + output
    scores2<<<cdiv(NCELL, BT), BT, 0, stream>>>(hs2, ht2, a1_o, a2_o, t1s, t1t, t2t, t2s);
    edge_max2<<<cdiv(NEDGE, BT), BT, 0, stream>>>(et, es, t1s, t1t, t2t, t2s, M1, M2);
    edge_acc2<<<cdiv(NEDGE * NC, BT), BT, 0, stream>>>(et, es, t1s, t1t, t2t, t2s,
                                                       M1, M2, hs2, ht2, D1, D2, A2ts, A2st);
    finalize2<<<cdiv(NCELL, BT), BT, 0, stream>>>(A2st, A2ts, D2, D1, out);
}